// GIN_91268055040640
// MI455X (gfx1250) — compile-verified
//
#include <hip/hip_runtime.h>

#define N_NODES    500000
#define N_EDGES    8000000
#define N_GRAPHS   4096
#define IN_DIM     9
#define HID        64
#define AGG_STRIDE 12            // pad 9 -> 12 so WMMA K-loop (step 4) sees zeros
#define K1         12            // padded K of layer 1
#define TILES      (N_NODES / 16)   // 31250 16-node tiles
#define WPB        8             // waves per block in the MLP kernel

typedef __attribute__((ext_vector_type(2))) float v2f;
typedef __attribute__((ext_vector_type(8))) float v8f;

// --------------------------------------------- zero scratch + build padded W1
__global__ __launch_bounds__(256) void gin_prep_kernel(
    const float* __restrict__ w1, float* __restrict__ ws_zero, int nzero,
    float* __restrict__ w1p) {
    int i = blockIdx.x * blockDim.x + threadIdx.x;
    if (i < nzero) ws_zero[i] = 0.0f;
    if (i < K1 * HID) {
        int r = i >> 6, c = i & 63;
        w1p[i] = (r < IN_DIM) ? w1[r * HID + c] : 0.0f;
    }
}

// ------------------------------------------- edge scatter: agg[dst] += x[src]
__global__ __launch_bounds__(256) void gin_edge_scatter_kernel(
    const float* __restrict__ x, const int* __restrict__ eidx,
    float* __restrict__ agg) {
    int e = blockIdx.x * blockDim.x + threadIdx.x;
    if (e >= N_EDGES) return;
    int s = eidx[e];
    int d = eidx[N_EDGES + e];
    const float* xr = x + (long)s * IN_DIM;
    float*       ar = agg + (long)d * AGG_STRIDE;
#pragma unroll
    for (int k = 0; k < IN_DIM; ++k)
        unsafeAtomicAdd(&ar[k], xr[k]);
}

// ------------- fused MLP (WMMA f32 16x16x4) + global_add_pool
// 8 waves / block; one 16-node tile per wave; W1p/W2 staged in LDS once per
// block (W2 via async global->LDS DMA, ASYNCcnt-tracked).
__global__ __launch_bounds__(256) void gin_mlp_pool_kernel(
    const float* __restrict__ x,   const float* __restrict__ agg,
    const int*   __restrict__ batch,
    const float* __restrict__ w1p, const float* __restrict__ b1,
    const float* __restrict__ w2,  const float* __restrict__ b2,
    float* __restrict__ pooled) {
    __shared__ float w2s[HID * HID];            // 16 KB, shared by all 8 waves
    __shared__ float w1s[K1 * HID];             // 3 KB padded W1
    __shared__ float lds_x[WPB][16 * AGG_STRIDE];
    __shared__ float lds_h[WPB][16 * HID];

    const int tid  = threadIdx.x;
    const int wv   = tid >> 5;
    const int lane = tid & 31;

    const int  tileIdx = blockIdx.x * WPB + wv;
    const bool active  = tileIdx < TILES;
    const int  tile    = active ? tileIdx : (TILES - 1);   // clamp: keep barriers uniform
    const int  node0   = tile * 16;
    const int  mrow    = lane & 15;
    const int  khalf   = (lane >> 4) << 1;      // lanes 16-31 carry K+2,K+3

    // ---- Stage W2 into LDS with async global->LDS B128 DMA (512 B/inst/wave).
    {
        unsigned lbase = (unsigned)(unsigned long long)(void*)w2s; // LDS byte offset
        const char* gbase = (const char*)w2;
#pragma unroll
        for (int j = 0; j < 4; ++j) {
            unsigned off  = (unsigned)(((wv * 4 + j) * 32 + lane) * 16);
            unsigned ldsa = lbase + off;
            asm volatile("global_load_async_to_lds_b128 %0, %1, %2"
                         :: "v"(ldsa), "v"(off), "s"(gbase) : "memory");
        }
    }

    // ---- Stage padded W1 (branchless; already zero-padded in workspace).
    for (int i = tid; i < K1 * HID; i += 256) w1s[i] = w1p[i];

    // ---- Stage this wave's (agg + x) tile, zero-padded to 12 columns.
    for (int i = lane; i < 16 * AGG_STRIDE; i += 32) {
        int r = i / AGG_STRIDE, c = i % AGG_STRIDE;
        float v = 0.0f;
        if (c < IN_DIM)
            v = agg[(long)(node0 + r) * AGG_STRIDE + c]
              + x  [(long)(node0 + r) * IN_DIM     + c];
        lds_x[wv][i] = v;
    }

    asm volatile("s_wait_asynccnt 0x0" ::: "memory");   // our async DMAs done
    __syncthreads();                                    // all waves' staging visible

    // ---- Layer 1: [16x12] @ [12x64] + b1, four 16-wide N tiles
    v8f acc[4];
#pragma unroll
    for (int t = 0; t < 4; ++t) {
        float bv = b1[t * 16 + mrow];
#pragma unroll
        for (int r = 0; r < 8; ++r) acc[t][r] = bv;
    }
#pragma unroll
    for (int kk = 0; kk < K1; kk += 4) {
        v2f a;
        a.x = lds_x[wv][mrow * AGG_STRIDE + kk + khalf + 0];
        a.y = lds_x[wv][mrow * AGG_STRIDE + kk + khalf + 1];
#pragma unroll
        for (int t = 0; t < 4; ++t) {
            int n = t * 16 + mrow;
            v2f b;
            b.x = w1s[(kk + khalf + 0) * HID + n];
            b.y = w1s[(kk + khalf + 1) * HID + n];
            acc[t] = __builtin_amdgcn_wmma_f32_16x16x4_f32(
                false, a, false, b, (short)0, acc[t], false, false);
        }
    }

    // ReLU, store hidden tile row-major to LDS (D-layout -> A-layout hop).
#pragma unroll
    for (int t = 0; t < 4; ++t) {
#pragma unroll
        for (int r = 0; r < 8; ++r) {
            int m = r + ((lane >> 4) << 3);
            int n = t * 16 + mrow;
            lds_h[wv][m * HID + n] = fmaxf(acc[t][r], 0.0f);
        }
    }
    __syncthreads();

    // ---- Layer 2: [16x64] @ [64x64] + b2 (W2 from LDS)
    v8f acc2[4];
#pragma unroll
    for (int t = 0; t < 4; ++t) {
        float bv = b2[t * 16 + mrow];
#pragma unroll
        for (int r = 0; r < 8; ++r) acc2[t][r] = bv;
    }
#pragma unroll
    for (int kk = 0; kk < HID; kk += 4) {
        v2f a;
        a.x = lds_h[wv][mrow * HID + kk + khalf + 0];
        a.y = lds_h[wv][mrow * HID + kk + khalf + 1];
#pragma unroll
        for (int t = 0; t < 4; ++t) {
            int n = t * 16 + mrow;
            v2f b;
            b.x = w2s[(kk + khalf + 0) * HID + n];
            b.y = w2s[(kk + khalf + 1) * HID + n];
            acc2[t] = __builtin_amdgcn_wmma_f32_16x16x4_f32(
                false, a, false, b, (short)0, acc2[t], false, false);
        }
    }

    // ---- global_add_pool. batch is sorted: uniform tiles take the reduced path.
    if (active) {
        int g0  = batch[node0];
        int g15 = batch[node0 + 15];
        if (g0 == g15) {
#pragma unroll
            for (int t = 0; t < 4; ++t) {
                float s = 0.0f;
#pragma unroll
                for (int r = 0; r < 8; ++r) s += acc2[t][r];
                s += __shfl_xor(s, 16, 32);        // add the M=8..15 half
                if (lane < 16)
                    unsafeAtomicAdd(&pooled[g0 * HID + t * 16 + lane], s);
            }
        } else {
#pragma unroll
            for (int t = 0; t < 4; ++t) {
                int n = t * 16 + mrow;
#pragma unroll
                for (int r = 0; r < 8; ++r) {
                    int m = r + ((lane >> 4) << 3);
                    int g = batch[node0 + m];
                    unsafeAtomicAdd(&pooled[g * HID + n], acc2[t][r]);
                }
            }
        }
    }
}

// ---------------------------------------------------------------- readout 64->1
__global__ __launch_bounds__(256) void gin_readout_kernel(
    const float* __restrict__ pooled, const float* __restrict__ w3,
    const float* __restrict__ b3, float* __restrict__ out) {
    int g = blockIdx.x * blockDim.x + threadIdx.x;
    if (g >= N_GRAPHS) return;
    float s = b3[0];
#pragma unroll 8
    for (int k = 0; k < HID; ++k) s += pooled[g * HID + k] * w3[k];
    out[g] = s;
}

extern "C" void kernel_launch(void* const* d_in, const int* in_sizes, int n_in,
                              void* d_out, int out_size, void* d_ws, size_t ws_size,
                              hipStream_t stream) {
    const float* x     = (const float*)d_in[0];
    const int*   eidx  = (const int*)  d_in[1];
    const int*   batch = (const int*)  d_in[2];
    const float* w1    = (const float*)d_in[3];
    const float* b1    = (const float*)d_in[4];
    const float* w2    = (const float*)d_in[5];
    const float* b2    = (const float*)d_in[6];
    const float* w3    = (const float*)d_in[7];
    const float* b3    = (const float*)d_in[8];
    float* out = (float*)d_out;

    float* agg    = (float*)d_ws;                        // [N_NODES, 12]
    float* pooled = agg + (size_t)N_NODES * AGG_STRIDE;  // [N_GRAPHS, 64]
    float* w1p    = pooled + (size_t)N_GRAPHS * HID;     // [12, 64] zero-padded W1

    const int nzero = N_NODES * AGG_STRIDE + N_GRAPHS * HID;
    gin_prep_kernel<<<(nzero + 255) / 256, 256, 0, stream>>>(
        w1, (float*)d_ws, nzero, w1p);

    gin_edge_scatter_kernel<<<(N_EDGES + 255) / 256, 256, 0, stream>>>(x, eidx, agg);

    gin_mlp_pool_kernel<<<(TILES + WPB - 1) / WPB, 32 * WPB, 0, stream>>>(
        x, agg, batch, w1p, b1, w2, b2, pooled);

    gin_readout_kernel<<<(N_GRAPHS + 255) / 256, 256, 0, stream>>>(
        pooled, w3, b3, out);
}